// QuantumActor_24764781428766
// MI455X (gfx1250) — compile-verified
//
#include <hip/hip_runtime.h>
#include <math.h>

typedef float v2f __attribute__((ext_vector_type(2)));
typedef float v8f __attribute__((ext_vector_type(8)));

__device__ __forceinline__ float2 cmul(float2 a, float2 b) {
    return make_float2(a.x * b.x - a.y * b.y, a.x * b.y + a.y * b.x);
}

// ---------------------------------------------------------------------------
// Kernel 1: build compressed 32x16 real matrix Mt = [Vr; Vi] where
// V[:,i] = U[:,i] (popcount(i) even) or i*U[:,i] (odd), U = fixed 16x16
// unitary of the weight-parameterized circuit part (2 layers of CNOT ring +
// RZ,RY,RZ per wire).  Then psi_out = V y with y[16] the per-sample signed
// magnitudes of the product-state encoding.  One thread per basis column.
// Wire w <-> bit mask (8 >> w): wire 0 is the MSB (ref axis order).
// ---------------------------------------------------------------------------
__global__ void build_gate_matrix(const float* __restrict__ wt,
                                  float* __restrict__ mt) {
    const int col = threadIdx.x;
    if (col >= 16) return;

    float2 psi[16];
#pragma unroll
    for (int i = 0; i < 16; ++i)
        psi[i] = make_float2(i == col ? 1.f : 0.f, 0.f);

#pragma unroll
    for (int j = 0; j < 2; ++j) {
        // CNOT ring (0,1),(1,2),(2,3),(3,0): control bit 1 -> flip target bit
        const int cws[4] = {0, 1, 2, 3};
        const int tws[4] = {1, 2, 3, 0};
#pragma unroll
        for (int g = 0; g < 4; ++g) {
            const int mc = 8 >> cws[g], mtk = 8 >> tws[g];
#pragma unroll
            for (int i = 0; i < 16; ++i) {
                if ((i & mc) && !(i & mtk)) {
                    float2 tmp = psi[i];
                    psi[i] = psi[i | mtk];
                    psi[i | mtk] = tmp;
                }
            }
        }
        // per-wire RZ(t0), RY(t1), RZ(t2); weights layout (4,2,3) row-major
#pragma unroll
        for (int q = 0; q < 4; ++q) {
            const int m = 8 >> q;
            const float t0 = wt[q * 6 + j * 3 + 0];
            const float t1 = wt[q * 6 + j * 3 + 1];
            const float t2 = wt[q * 6 + j * 3 + 2];
            {   // RZ(t0): bit0 *= (c,-s); bit1 *= (c,+s)
                float sh, ch;
                sincosf(0.5f * t0, &sh, &ch);
#pragma unroll
                for (int i = 0; i < 16; ++i)
                    psi[i] = cmul(psi[i], make_float2(ch, (i & m) ? sh : -sh));
            }
            {   // RY(t1): [[c,-s],[s,c]] (real)
                float sh, ch;
                sincosf(0.5f * t1, &sh, &ch);
#pragma unroll
                for (int i = 0; i < 16; ++i) {
                    if (!(i & m)) {
                        float2 a = psi[i], b = psi[i | m];
                        psi[i]     = make_float2(ch * a.x - sh * b.x, ch * a.y - sh * b.y);
                        psi[i | m] = make_float2(sh * a.x + ch * b.x, sh * a.y + ch * b.y);
                    }
                }
            }
            {   // RZ(t2)
                float sh, ch;
                sincosf(0.5f * t2, &sh, &ch);
#pragma unroll
                for (int i = 0; i < 16; ++i)
                    psi[i] = cmul(psi[i], make_float2(ch, (i & m) ? sh : -sh));
            }
        }
    }

    // Mt is 32x16 row-major; column col.  Odd parity: i*U = -Ui + i*Ur.
    const int par = __popc(col) & 1;
#pragma unroll
    for (int r = 0; r < 16; ++r) {
        mt[r * 16 + col]        = par ? -psi[r].y : psi[r].x;   // Vr
        mt[(r + 16) * 16 + col] = par ?  psi[r].x : psi[r].y;   // Vi
    }
}

// ---------------------------------------------------------------------------
// Kernel 2: per wave, 16-sample tile.  D[comp, samp] = Mt * Y^T via
// 8x V_WMMA_F32_16X16X4_F32 (2 component tiles: Re/Im, K=16 in 4 chunks).
// Each lane then holds 8 |amp|^2 of ONE sample in its D registers ->
// in-lane reduction + one shfl_xor(16), coalesced float2 store.
// ---------------------------------------------------------------------------
__global__ void __launch_bounds__(256)
qcirc_wmma(const float* __restrict__ state,
           const float* __restrict__ mt,
           float* __restrict__ out, int ntiles) {
    const int lane = threadIdx.x & 31;
    const int n    = lane & 15;   // A row / B,D column within half
    const int h    = lane >> 4;   // half: K pair (A/B), D row group
    const int wavesPerBlock = blockDim.x >> 5;
    const int waveId = blockIdx.x * wavesPerBlock + (threadIdx.x >> 5);
    const int nWaves = gridDim.x * wavesPerBlock;

    // A operand (constant): af[t][k] = Mt[16t+n, 4k+2h .. +1]
    v2f af[2][4];
#pragma unroll
    for (int t = 0; t < 2; ++t) {
        const float2* row = (const float2*)(mt + (t * 16 + n) * 16);
#pragma unroll
        for (int k = 0; k < 4; ++k) {
            float2 v = row[2 * k + h];
            af[t][k][0] = v.x;
            af[t][k][1] = v.y;
        }
    }

    for (int tile = waveId; tile < ntiles; tile += nWaves) {
        const int samp = tile * 16 + n;          // lanes l and l+16: same sample
        const float4 st = ((const float4*)state)[samp];

        // half-angles per wire; atan wires trig-free: cos(atan x)=rsqrt(1+x^2).
        // args are small finite values -> fast hw sin/cos path is fine.
        float s0, c0, s2, c2;
        __sincosf(0.32724923f * st.x, &s0, &c0);  // (pi/4.8)/2 * x
        __sincosf(3.75788634f * st.z, &s2, &c2);  // (pi/0.418)/2 * z
        const float r1 = rsqrtf(1.f + st.y * st.y);
        const float c1 = r1, s1 = st.y * r1;
        const float r3 = rsqrtf(1.f + st.w * st.w);
        const float c3 = r3, s3 = st.w * r3;

        // y_i = sign(p)*prod(c/s), p = popcount(i); this lane: i2 = h.
        // sign negative iff ((p+1)&2).
        const float m0v[2] = {c0, s0};
        const float m1v[2] = {c1, s1};
        const float m3v[2] = {c3, s3};
        const float m2 = h ? s2 : c2;

        v2f yb[4];   // B operand: yb[k] = {y[4k+2h], y[4k+2h+1]} local entries
#pragma unroll
        for (int i0 = 0; i0 < 2; ++i0) {
#pragma unroll
            for (int i1 = 0; i1 < 2; ++i1) {
                const float g = m0v[i0] * m1v[i1] * m2;
                const int k = i0 * 2 + i1;       // chunk index (i0=k>>1,i1=k&1)
#pragma unroll
                for (int i3 = 0; i3 < 2; ++i3) {
                    const float v = g * m3v[i3];
                    const int p0 = i0 + i1 + i3;          // + h at runtime
                    const bool neg = (((p0 + h + 1) & 2) != 0);
                    yb[k][i3] = neg ? -v : v;
                }
            }
        }

        v8f acc0 = {};  // Re(psi_out[comp])  comp = r + 8h
        v8f acc1 = {};  // Im(psi_out[comp])
#pragma unroll
        for (int k = 0; k < 4; ++k) {
            acc0 = __builtin_amdgcn_wmma_f32_16x16x4_f32(
                false, af[0][k], false, yb[k], (short)0, acc0, false, false);
            acc1 = __builtin_amdgcn_wmma_f32_16x16x4_f32(
                false, af[1][k], false, yb[k], (short)0, acc1, false, false);
        }

        // p[r] = |psi_out[r+8h]|^2 of sample n.  z0 sign: amp bit3 = h;
        // z1 sign: amp bit2 = r&4.
        float sA = 0.f, sB = 0.f;
#pragma unroll
        for (int r = 0; r < 4; ++r)
            sA += acc0[r] * acc0[r] + acc1[r] * acc1[r];
#pragma unroll
        for (int r = 4; r < 8; ++r)
            sB += acc0[r] * acc0[r] + acc1[r] * acc1[r];

        float z0p = h ? -(sA + sB) : (sA + sB);
        float z1p = sA - sB;
        const float z0 = z0p + __shfl_xor(z0p, 16, 32);
        const float z1 = z1p + __shfl_xor(z1p, 16, 32);

        // softmax([(z0+1)/2,(z1+1)/2]) = [sigmoid((z0-z1)/2), 1-...]
        if (h == 0) {
            const float d = 0.5f * (z0 - z1);
            const float e = 1.f / (1.f + __expf(-d));
            ((float2*)out)[tile * 16 + n] = make_float2(e, 1.f - e);
        }
    }
}

extern "C" void kernel_launch(void* const* d_in, const int* in_sizes, int n_in,
                              void* d_out, int out_size, void* d_ws, size_t ws_size,
                              hipStream_t stream) {
    (void)n_in; (void)out_size; (void)ws_size;
    const float* state   = (const float*)d_in[0];   // [B,4] f32
    const float* weights = (const float*)d_in[1];   // [4,2,3] f32
    float* out = (float*)d_out;                     // [B,2] f32
    float* mt  = (float*)d_ws;                      // 32x16 f32 = 2 KB

    const int B = in_sizes[0] / 4;
    const int ntiles = B / 16;

    build_gate_matrix<<<1, 32, 0, stream>>>(weights, mt);

    const int threads = 256;                        // 8 waves/block
    int blocks = 1024;                              // grid-stride over tiles
    const int maxBlocks = (ntiles + (threads / 32) - 1) / (threads / 32);
    if (blocks > maxBlocks) blocks = maxBlocks;
    if (blocks < 1) blocks = 1;
    qcirc_wmma<<<blocks, threads, 0, stream>>>(state, mt, out, ntiles);
}